// ResNet_3G_365_37194416783940
// MI455X (gfx1250) — compile-verified
//
#include <hip/hip_runtime.h>
#include <hip/hip_bf16.h>
#include <math.h>

typedef float v2f __attribute__((ext_vector_type(2)));
typedef float v8f __attribute__((ext_vector_type(8)));

#define DIM 256
#define HW 196
#define KPAD 224
#define TRIU_LEN 32896
#define NCLS 365

// ---------------------------------------------------------------------------
// 1) Center x over spatial dim, emit xc [256 x 224] (K zero-padded) and
//    xcT [224 x 256] per batch so the covariance is a plain NN GEMM.
// ---------------------------------------------------------------------------
__global__ __launch_bounds__(256) void prep_xc_kernel(const float* __restrict__ x,
                                                      float* __restrict__ xc,
                                                      float* __restrict__ xcT) {
    const int b = blockIdx.x;
    const int d = threadIdx.x;               // 256 threads = 256 channels
    const float* xr = x + ((size_t)b * DIM + d) * HW;
    float s = 0.f;
    for (int m = 0; m < HW; ++m) s += xr[m];
    const float mean = s * (1.0f / (float)HW);
    float* xcr  = xc  + ((size_t)b * DIM + d) * KPAD;
    float* xcTb = xcT + (size_t)b * KPAD * DIM;
    for (int m = 0; m < HW; ++m) {
        const float v = xr[m] - mean;
        xcr[m] = v;
        xcTb[(size_t)m * DIM + d] = v;
    }
    for (int m = HW; m < KPAD; ++m) {
        xcr[m] = 0.f;
        xcTb[(size_t)m * DIM + d] = 0.f;
    }
}

// ---------------------------------------------------------------------------
// 2) Batched GEMM: C[b] = alpha * (A[b] @ B[b]) + beta * I
//    A: [256 x K] row-major, row stride lda, batch stride 256*lda
//    B: [K x 256] row-major, row stride 256, batch stride K*256
//    C: [256 x 256]
//    Block = 256 threads (8 waves) -> 128x128 C tile; grid (2, 2, 64).
//    Waves form a 4x2 grid; each wave owns a 32x64 sub-block = 2x4 WMMA tiles
//    (2 A-frag + 4 B-frag loads per 8 WMMAs).  K staged in 32-wide chunks.
//    LDS layouts make every WMMA operand ONE aligned ds_load_b64:
//      As[row][k]  (stride 34, even -> 8B-aligned (k,k+1) pairs)
//      Bt[col][k]  (B transposed; (k,k+1) at fixed col contiguous)
// ---------------------------------------------------------------------------
__global__ __launch_bounds__(256) void gemm256_kernel(const float* __restrict__ A,
                                                      const float* __restrict__ B,
                                                      float* __restrict__ C,
                                                      int K, int lda,
                                                      float alpha, float beta) {
    __shared__ __align__(16) float As[128][34];   // 17,408 B
    __shared__ __align__(16) float Bt[128][34];   // 17,408 B (B chunk, transposed)

    const int tid  = threadIdx.x;
    const int lane = tid & 31;
    const int wave = tid >> 5;
    const int half = lane >> 4;      // 0: lanes 0-15, 1: lanes 16-31
    const int l16  = lane & 15;
    const int wr   = wave & 3;       // wave-row 0..3  (32 rows each)
    const int wc   = wave >> 2;      // wave-col 0..1  (64 cols each)

    const int rowBase = blockIdx.y * 128;
    const int colBase = blockIdx.x * 128;
    const int b = blockIdx.z;

    const float* Ab = A + (size_t)b * DIM * (size_t)lda;
    const float* Bb = B + (size_t)b * (size_t)K * DIM;
    float*       Cb = C + (size_t)b * DIM * DIM;

    v8f acc[2][4];
#pragma unroll
    for (int s = 0; s < 2; ++s)
#pragma unroll
        for (int t = 0; t < 4; ++t) acc[s][t] = (v8f){0.f,0.f,0.f,0.f,0.f,0.f,0.f,0.f};

    for (int k0 = 0; k0 < K; k0 += 32) {
        // --- stage A chunk: 128 rows x 32 k (1024 float4, 4 per thread) ---
#pragma unroll
        for (int i = 0; i < 4; ++i) {
            const int idx = tid + i * 256;       // 0..1023
            const int r   = idx >> 3;            // 0..127
            const int kq  = idx & 7;             // 0..7 (float4 within row)
            const float4 v = *(const float4*)(Ab + (size_t)(rowBase + r) * lda + k0 + kq * 4);
            *(v2f*)&As[r][kq * 4 + 0] = (v2f){v.x, v.y};   // 8B-aligned ds_store_b64
            *(v2f*)&As[r][kq * 4 + 2] = (v2f){v.z, v.w};
        }
        // --- stage B chunk transposed: 32 k x 128 cols -> Bt[col][k] ---
#pragma unroll
        for (int i = 0; i < 4; ++i) {
            const int idx = tid + i * 256;       // 0..1023
            const int kr  = idx >> 5;            // 0..31
            const int cq  = idx & 31;            // 0..31 (float4 within row)
            const float4 v = *(const float4*)(Bb + (size_t)(k0 + kr) * DIM + colBase + cq * 4);
            Bt[cq * 4 + 0][kr] = v.x;
            Bt[cq * 4 + 1][kr] = v.y;
            Bt[cq * 4 + 2][kr] = v.z;
            Bt[cq * 4 + 3][kr] = v.w;
        }
        __syncthreads();

#pragma unroll
        for (int kk = 0; kk < 32; kk += 4) {
            // Preload all fragments for this k-step so DS issues batch up and
            // overlap with the WMMA burst (waits on nonzero dscnt).
            v2f afr[2];
#pragma unroll
            for (int s = 0; s < 2; ++s)
                afr[s] = *(const v2f*)&As[wr * 32 + s * 16 + l16][kk + 2 * half];
            v2f bfr[4];
#pragma unroll
            for (int t = 0; t < 4; ++t)
                bfr[t] = *(const v2f*)&Bt[wc * 64 + t * 16 + l16][kk + 2 * half];
#pragma unroll
            for (int s = 0; s < 2; ++s)
#pragma unroll
                for (int t = 0; t < 4; ++t)
                    acc[s][t] = __builtin_amdgcn_wmma_f32_16x16x4_f32(
                        /*neg_a=*/false, afr[s], /*neg_b=*/false, bfr[t],
                        /*c_mod=*/(short)0, acc[s][t],
                        /*reuse_a=*/false, /*reuse_b=*/false);
        }
        __syncthreads();
    }

    // --- epilogue: C = alpha*acc + beta*I ---
#pragma unroll
    for (int s = 0; s < 2; ++s) {
        const int mBase = rowBase + wr * 32 + s * 16;
#pragma unroll
        for (int t = 0; t < 4; ++t) {
            const int n = colBase + wc * 64 + t * 16 + l16;
#pragma unroll
            for (int v = 0; v < 8; ++v) {
                const int m = mBase + v + 8 * half;  // C layout: VGPR v -> M = v + 8*half
                float val = alpha * acc[s][t][v];
                if (m == n) val += beta;
                Cb[(size_t)m * DIM + n] = val;
            }
        }
    }
}

// ---------------------------------------------------------------------------
// 3) Per-batch: normA = trace(cov); Y = cov / normA (in place); Z = I.
// ---------------------------------------------------------------------------
__global__ __launch_bounds__(256) void init_yz_kernel(float* __restrict__ Y,
                                                      float* __restrict__ Z,
                                                      float* __restrict__ normA) {
    const int b = blockIdx.x;
    const int t = threadIdx.x;
    float* Yb = Y + (size_t)b * DIM * DIM;
    float* Zb = Z + (size_t)b * DIM * DIM;
    __shared__ float red[256];
    red[t] = Yb[(size_t)t * DIM + t];
    __syncthreads();
    for (int s = 128; s > 0; s >>= 1) {
        if (t < s) red[t] += red[t + s];
        __syncthreads();
    }
    const float tr = red[0];
    if (t == 0) normA[b] = tr;
    const float inv = 1.0f / tr;
    for (int c = 0; c < DIM; ++c) Yb[(size_t)t * DIM + c] *= inv;
    for (int c = 0; c < DIM; ++c) Zb[(size_t)t * DIM + c] = (c == t) ? 1.0f : 0.0f;
}

// ---------------------------------------------------------------------------
// 4) Gather lower-tri entries (row-major, c<=r) with the sqrt(normA) rescale.
//    grid (256 rows, 64 batch), block 256.
// ---------------------------------------------------------------------------
__global__ __launch_bounds__(256) void gather_triu_kernel(const float* __restrict__ Y,
                                                          const float* __restrict__ normA,
                                                          float* __restrict__ triu) {
    const int r = blockIdx.x;
    const int b = blockIdx.y;
    const int c = threadIdx.x;
    if (c <= r) {
        const float s = sqrtf(normA[b]);
        triu[(size_t)b * TRIU_LEN + (size_t)r * (r + 1) / 2 + c] =
            Y[(size_t)b * DIM * DIM + (size_t)r * DIM + c] * s;
    }
}

// ---------------------------------------------------------------------------
// 5) FC: out[64 x 365] = triu[64 x 32896] @ W^T + bias.  K = 257 * 128 exact.
//    grid 46 blocks (8 classes each), block 256 threads, LDS K-tiling so W is
//    read once and triu stays L2-resident.
// ---------------------------------------------------------------------------
__global__ __launch_bounds__(256) void fc_kernel(const float* __restrict__ T,
                                                 const float* __restrict__ W,
                                                 const float* __restrict__ bias,
                                                 float* __restrict__ out) {
    __shared__ float tS[64][132];
    __shared__ float wS[8][132];
    const int tid   = threadIdx.x;
    const int nBase = blockIdx.x * 8;

    const int o0 = tid, o1 = tid + 256;          // 512 outputs = 64 b x 8 n
    const int b0 = o0 >> 3, n0 = o0 & 7;
    const int b1 = o1 >> 3, n1 = o1 & 7;
    float acc0 = 0.f, acc1 = 0.f;

    for (int k0 = 0; k0 < TRIU_LEN; k0 += 128) {
        // stage triu chunk: 64 x 128 = 2048 float4, 8 per thread
#pragma unroll
        for (int i = 0; i < 8; ++i) {
            const int idx = tid + i * 256;       // 0..2047
            const int r   = idx >> 5;            // 0..63
            const int cq  = idx & 31;            // 0..31
            const float4 v = *(const float4*)(T + (size_t)r * TRIU_LEN + k0 + cq * 4);
            tS[r][cq * 4 + 0] = v.x;
            tS[r][cq * 4 + 1] = v.y;
            tS[r][cq * 4 + 2] = v.z;
            tS[r][cq * 4 + 3] = v.w;
        }
        // stage W chunk: 8 x 128 = 256 float4, 1 per thread
        {
            const int r  = tid >> 5;             // 0..7
            const int cq = tid & 31;
            const int n  = nBase + r;
            float4 v = make_float4(0.f, 0.f, 0.f, 0.f);
            if (n < NCLS) v = *(const float4*)(W + (size_t)n * TRIU_LEN + k0 + cq * 4);
            wS[r][cq * 4 + 0] = v.x;
            wS[r][cq * 4 + 1] = v.y;
            wS[r][cq * 4 + 2] = v.z;
            wS[r][cq * 4 + 3] = v.w;
        }
        __syncthreads();
#pragma unroll 4
        for (int kk = 0; kk < 128; ++kk) {
            acc0 += tS[b0][kk] * wS[n0][kk];     // same-b lanes broadcast
            acc1 += tS[b1][kk] * wS[n1][kk];
        }
        __syncthreads();
    }
    const int gn0 = nBase + n0, gn1 = nBase + n1;
    if (gn0 < NCLS) out[(size_t)b0 * NCLS + gn0] = acc0 + bias[gn0];
    if (gn1 < NCLS) out[(size_t)b1 * NCLS + gn1] = acc1 + bias[gn1];
}

// ---------------------------------------------------------------------------
// Host-side orchestration
// ---------------------------------------------------------------------------
extern "C" void kernel_launch(void* const* d_in, const int* in_sizes, int n_in,
                              void* d_out, int out_size, void* d_ws, size_t ws_size,
                              hipStream_t stream) {
    const float* x    = (const float*)d_in[0];   // [64, 256, 14, 14]
    const float* fc_w = (const float*)d_in[1];   // [365, 32896]
    const float* fc_b = (const float*)d_in[2];   // [365]
    float* out = (float*)d_out;                  // [64, 365]

    float* wsf = (float*)d_ws;
    const size_t BUF = (size_t)64 * DIM * DIM;   // 4,194,304 floats per matrix buffer
    float* bA    = wsf + 0 * BUF;                // Y ping
    float* bB    = wsf + 1 * BUF;                // Z ping
    float* bC    = wsf + 2 * BUF;                // T
    float* bD    = wsf + 3 * BUF;                // xc   -> Y pong
    float* bE    = wsf + 4 * BUF;                // xcT  -> Z pong
    float* triu  = wsf + 5 * BUF;                // [64, 32896]
    float* normA = triu + (size_t)64 * TRIU_LEN; // [64]

    // 1) center + transpose staging
    prep_xc_kernel<<<64, 256, 0, stream>>>(x, bD, bE);

    // 2) cov: Y = (xc @ xcT)/196 + 1e-7 I   (zero-padded K contributes 0)
    gemm256_kernel<<<dim3(2, 2, 64), 256, 0, stream>>>(
        bD, bE, bA, KPAD, KPAD, 1.0f / (float)HW, 1e-7f);

    // 3) normA = trace; Y /= normA; Z = I
    init_yz_kernel<<<64, 256, 0, stream>>>(bA, bB, normA);

    // 4) Newton-Schulz x5:
    //    T = 1.5 I - 0.5 Z@Y ; Y' = Y@T ; Z' = T@Z
    float* Y = bA; float* Z = bB;
    float* Yn = bD; float* Zn = bE;
    for (int it = 0; it < 5; ++it) {
        gemm256_kernel<<<dim3(2, 2, 64), 256, 0, stream>>>(Z, Y, bC, DIM, DIM, -0.5f, 1.5f);
        gemm256_kernel<<<dim3(2, 2, 64), 256, 0, stream>>>(Y, bC, Yn, DIM, DIM, 1.0f, 0.0f);
        gemm256_kernel<<<dim3(2, 2, 64), 256, 0, stream>>>(bC, Z, Zn, DIM, DIM, 1.0f, 0.0f);
        float* t;
        t = Y; Y = Yn; Yn = t;
        t = Z; Z = Zn; Zn = t;
    }

    // 5) tril gather with * sqrt(normA)
    gather_triu_kernel<<<dim3(256, 64), 256, 0, stream>>>(Y, normA, triu);

    // 6) FC
    fc_kernel<<<(NCLS + 7) / 8, 256, 0, stream>>>(triu, fc_w, fc_b, out);
}